// JacobianGluer_82171314307532
// MI455X (gfx1250) — compile-verified
//
#include <hip/hip_runtime.h>
#include <hip/hip_bf16.h>

// CDNA5 / gfx1250. wave32. V_WMMA_F32_16X16X4_F32 accumulates G = w @ F.

typedef __attribute__((ext_vector_type(2))) float v2f;
typedef __attribute__((ext_vector_type(8))) float v8f;

#define INV_R   0.4f                  // 1 / 2.5
#define LOG2E   1.4426950408889634f

#define NCOL   16                     // O*(D+1) = 4*4
#define SLICE  256                    // n-rows staged in LDS per outer iteration
#define WAVES  8
#define BLOCK  (WAVES * 32)
#define MTILE  (WAVES * 16)           // 128 m-rows per workgroup

// ---------------------------------------------------------------------------
// Precompute: pts4[n] = {p/r, |p/r|^2}   (pre-scaled so dist^2/r^2 is direct)
//             F[n, o*4+d] = J[n,o,d] (d<3);  F[n, o*4+3] = values[n,o] - p.J[n,o]
// ---------------------------------------------------------------------------
__global__ void __launch_bounds__(256)
jg_precompute(const float* __restrict__ points,
              const float* __restrict__ jac,
              const float* __restrict__ vals,
              float4* __restrict__ pts4,
              float*  __restrict__ F,
              int N) {
    int n = blockIdx.x * blockDim.x + threadIdx.x;
    if (n >= N) return;
    float px = points[n * 3 + 0];
    float py = points[n * 3 + 1];
    float pz = points[n * 3 + 2];
    float qx = px * INV_R, qy = py * INV_R, qz = pz * INV_R;
    pts4[n] = make_float4(qx, qy, qz, fmaf(qx, qx, fmaf(qy, qy, qz * qz)));
#pragma unroll
    for (int o = 0; o < 4; ++o) {
        float j0 = jac[(n * 4 + o) * 3 + 0];
        float j1 = jac[(n * 4 + o) * 3 + 1];
        float j2 = jac[(n * 4 + o) * 3 + 2];
        float c  = vals[n * 4 + o] - fmaf(px, j0, fmaf(py, j1, pz * j2));
        F[n * NCOL + o * 4 + 0] = j0;
        F[n * NCOL + o * 4 + 1] = j1;
        F[n * NCOL + o * 4 + 2] = j2;
        F[n * NCOL + o * 4 + 3] = c;
    }
}

// Bump weight on pre-scaled coords: r2 = |x' - p'|^2 (already /r^2).
// w = exp(1 - 1/(1-r2)) inside, 0 outside. Dead-branch inf is discarded by
// the final cndmask, so no "safe t" select is needed.
__device__ __forceinline__ float bump_w(float xnorm, float4 q,
                                        float x0, float x1, float x2) {
    float dot = fmaf(x0, q.x, fmaf(x1, q.y, x2 * q.z));
    float r2  = fmaf(-2.0f, dot, xnorm + q.w);
    r2 = fmaxf(r2, 0.0f);
    float t = 1.0f - r2;
    // exp(1 - 1/t) = exp2(log2e - log2e * rcp(t))
    float w = __builtin_amdgcn_exp2f(
        fmaf(-LOG2E, __builtin_amdgcn_rcpf(t), LOG2E));
    return (r2 < 1.0f) ? w : 0.0f;
}

// ---------------------------------------------------------------------------
// Main: per wave, a 16-row m-tile; loop over N in LDS-staged slices; for each
// 16-n chunk compute the 16x16 bump-weight tile directly in WMMA A-layout and
// accumulate G[16m x 16col] with v_wmma_f32_16x16x4_f32 (4 per chunk).
// ---------------------------------------------------------------------------
__global__ void __launch_bounds__(BLOCK)
jg_main(const float*  __restrict__ x,
        const float4* __restrict__ pts4,
        const float*  __restrict__ F,
        float* __restrict__ out,
        int M, int N) {
    __shared__ float4 sPts[SLICE];            //  4 KB
    __shared__ float  sF[SLICE * NCOL];       // 16 KB
    __shared__ float  sG[WAVES * 16 * 16];    //  8 KB

    const int tid    = threadIdx.x;
    const int lane   = tid & 31;
    const int wave   = tid >> 5;
    const int mlocal = lane & 15;             // A-matrix row M (and B-matrix col N)
    const int k0     = (lane >> 4) << 1;      // A/B K half: lanes 0-15 -> K0, 16-31 -> K2
    const int mbase  = blockIdx.x * MTILE + wave * 16;
    const int m      = mbase + mlocal;

    // Pre-scaled x for the weight computation.
    const float x0 = x[m * 3 + 0] * INV_R;
    const float x1 = x[m * 3 + 1] * INV_R;
    const float x2 = x[m * 3 + 2] * INV_R;
    const float xnorm = fmaf(x0, x0, fmaf(x1, x1, x2 * x2));

    v8f acc = {0.f, 0.f, 0.f, 0.f, 0.f, 0.f, 0.f, 0.f};

    const int nslices = N / SLICE;
    for (int s = 0; s < nslices; ++s) {
        const int nbase = s * SLICE;
        __syncthreads();                      // previous slice fully consumed
        // Cooperative stage of this slice into LDS.
        sPts[tid] = pts4[nbase + tid];
        {
            const float4* Fg  = (const float4*)(F + (size_t)nbase * NCOL);
            float4*       sF4 = (float4*)sF;
#pragma unroll
            for (int j = 0; j < 4; ++j)
                sF4[tid * 4 + j] = Fg[tid * 4 + j];
        }
        if (s + 1 < nslices)                  // global_prefetch_b8 of next F slice
            __builtin_prefetch(F + (size_t)(nbase + SLICE) * NCOL + tid * 16, 0, 0);
        __syncthreads();

#pragma unroll 4
        for (int c = 0; c < SLICE / 16; ++c) {
            const int nb = c * 16;
#pragma unroll
            for (int t = 0; t < 4; ++t) {
                const int row0 = nb + t * 4 + k0;     // this lane's K rows
                const float4 p0 = sPts[row0];
                const float4 p1 = sPts[row0 + 1];
                v2f a;                                 // A: 16x4 f32 weight tile
                a.x = bump_w(xnorm, p0, x0, x1, x2);
                a.y = bump_w(xnorm, p1, x0, x1, x2);
                v2f b;                                 // B: 4x16 f32 feature rows
                b.x = sF[row0 * NCOL + mlocal];
                b.y = sF[(row0 + 1) * NCOL + mlocal];
                acc = __builtin_amdgcn_wmma_f32_16x16x4_f32(
                    false, a, false, b, (short)0, acc, false, false);
            }
        }
    }

    // Epilogue: spill G tile (C/D layout: lane l, vgpr v -> M = v + 8*(l/16),
    // col = l%16) through LDS, then out[m,o] = G[m,4o:4o+3].x_m + G[m,4o+3].
    __syncthreads();
    float* g = sG + wave * 256;
#pragma unroll
    for (int v = 0; v < 8; ++v)
        g[(v + 8 * (lane >> 4)) * 16 + (lane & 15)] = acc[v];
    __syncthreads();

#pragma unroll
    for (int r = 0; r < 2; ++r) {
        const int idx = lane * 2 + r;   // 64 (m,o) pairs per wave
        const int ml  = idx >> 2;
        const int o   = idx & 3;
        const int mg  = mbase + ml;
        const float y0 = x[mg * 3 + 0];
        const float y1 = x[mg * 3 + 1];
        const float y2 = x[mg * 3 + 2];
        const float g0 = g[ml * 16 + o * 4 + 0];
        const float g1 = g[ml * 16 + o * 4 + 1];
        const float g2 = g[ml * 16 + o * 4 + 2];
        const float g3 = g[ml * 16 + o * 4 + 3];
        out[mg * 4 + o] = fmaf(g0, y0, fmaf(g1, y1, fmaf(g2, y2, g3)));
    }
}

extern "C" void kernel_launch(void* const* d_in, const int* in_sizes, int n_in,
                              void* d_out, int out_size, void* d_ws, size_t ws_size,
                              hipStream_t stream) {
    const float* x      = (const float*)d_in[0];   // [M,3]
    const float* points = (const float*)d_in[1];   // [N,3]
    const float* jac    = (const float*)d_in[2];   // [N,4,3]
    const float* vals   = (const float*)d_in[3];   // [N,4]
    float* out = (float*)d_out;                    // [M,4]

    const int M = in_sizes[0] / 3;
    const int N = in_sizes[1] / 3;

    // Workspace: F [N x 16] f32, then pre-scaled packed points [N] float4.
    float*  F    = (float*)d_ws;
    float4* pts4 = (float4*)(F + (size_t)N * NCOL);

    jg_precompute<<<(N + 255) / 256, 256, 0, stream>>>(points, jac, vals, pts4, F, N);
    jg_main<<<M / MTILE, BLOCK, 0, stream>>>(x, pts4, F, out, M, N);
}